// MyModel_73538430042583
// MI455X (gfx1250) — compile-verified
//
#include <hip/hip_runtime.h>

// Reference: p = fixpoint(ptr); out[i,:] = table[row_idx[p_i], :]
// d_in[0] = table (float32, V*D), d_in[1] = ptr (int32, N), d_in[2] = row_idx (int32, N)
// d_out   = out (float32, N*D), D = 64.
//
// MI455X strategy:
//  - phase 1: per-thread fix-point chase (ptr is 8 MB -> L2 resident),
//    resolved row shared through LDS (one chase per row, not 16).
//  - phase 2: CDNA5 async data movement. Per-lane GLOBAL_LOAD_ASYNC_TO_LDS_B128
//    gathers the data-dependent table rows into a 64 KB LDS stage, then
//    GLOBAL_STORE_ASYNC_FROM_LDS_B128 streams the block's contiguous 64 KB
//    output tile with a non-temporal hint (keep the 51 MB table in the 192 MB
//    L2; don't cache the 512 MB write-once output). Transfers bypass VGPRs
//    and are tracked with ASYNCcnt, freeing wave issue slots to hide the
//    latency-bound chases.

typedef float v4f __attribute__((ext_vector_type(4)));

#define EMB_D        64          // floats per row
#define D_VEC        (EMB_D / 4) // float4 per row = 16
#define ROWS_PER_BLK 256
#define BLK_THREADS  256         // 8 wave32 per block

__global__ __launch_bounds__(BLK_THREADS) void
coq_embed_resolve_gather_async(const float* __restrict__ table,
                               const int*   __restrict__ ptr,
                               const int*   __restrict__ row_idx,
                               float*       __restrict__ out,
                               int n)
{
    __shared__ int s_row[ROWS_PER_BLK];
    __shared__ v4f s_stage[ROWS_PER_BLK * D_VEC];   // 64 KB staging tile

    const int t    = threadIdx.x;
    const int base = blockIdx.x * ROWS_PER_BLK;

    // ---- Phase 1: resolve each node's representative leaf. ----
    // Leaves self-point; internal nodes point strictly earlier -> terminates.
    const int row = base + t;
    if (row < n) {
        int p = ptr[row];
        int q = ptr[p];
        while (q != p) {
            p = q;
            q = ptr[p];
        }
        s_row[t] = row_idx[p];
    }
    __syncthreads();

    const v4f* __restrict__ tbl4 = (const v4f*)table;
    v4f*       __restrict__ out4 = (v4f*)out;

    // ---- Phase 2a: async gather table rows -> LDS (b128 per lane). ----
    // flat float4 index f = it*256 + t; local row r = f/16, column c = f%16.
    // 16 consecutive lanes fetch one contiguous 256 B table row.
    #pragma unroll
    for (int it = 0; it < 16; ++it) {
        const int f = it * BLK_THREADS + t;
        const int r = f >> 4;
        if (base + r < n) {
            const v4f* src = tbl4 + ((long long)s_row[r] * D_VEC + (f & 15));
            const unsigned lds = (unsigned)(size_t)&s_stage[f]; // AS(3) offset
            asm volatile("global_load_async_to_lds_b128 %0, %1, off"
                         :: "v"(lds), "v"(src)
                         : "memory");
        }
    }
    // Each lane's stores below read only LDS bytes its own loads wrote,
    // so waiting out this wave's ASYNCcnt is sufficient (no barrier needed).
    asm volatile("s_wait_asynccnt 0x0" ::: "memory");

    // ---- Phase 2b: async store LDS -> out, contiguous, non-temporal. ----
    // Output tile of this block is contiguous: dst_f4 = base*16 + f.
    #pragma unroll
    for (int it = 0; it < 16; ++it) {
        const int f = it * BLK_THREADS + t;
        const int r = f >> 4;
        if (base + r < n) {
            v4f* dst = out4 + ((long long)base * D_VEC + f);
            const unsigned lds = (unsigned)(size_t)&s_stage[f];
            asm volatile("global_store_async_from_lds_b128 %0, %1, off th:TH_STORE_NT"
                         :: "v"(dst), "v"(lds)
                         : "memory");
        }
    }
    // Drain our transfers before wave teardown (S_ENDPGM also waits-idle,
    // but make the dependency explicit).
    asm volatile("s_wait_asynccnt 0x0" ::: "memory");
}

extern "C" void kernel_launch(void* const* d_in, const int* in_sizes, int n_in,
                              void* d_out, int out_size, void* d_ws, size_t ws_size,
                              hipStream_t stream) {
    const float* table   = (const float*)d_in[0];
    const int*   ptr     = (const int*)  d_in[1];
    const int*   row_idx = (const int*)  d_in[2];
    float*       out     = (float*)      d_out;

    const int n    = in_sizes[1];                       // N nodes
    const int grid = (n + ROWS_PER_BLK - 1) / ROWS_PER_BLK;

    coq_embed_resolve_gather_async<<<grid, BLK_THREADS, 0, stream>>>(
        table, ptr, row_idx, out, n);
}